// Three_Bodies_RNN_54314156425432
// MI455X (gfx1250) — compile-verified
//
#include <hip/hip_runtime.h>
#include <math.h>

// Problem constants
#define B_   16
#define S_   1024
#define IN_  64
#define HID_ 256
#define OUT_ 64

typedef __attribute__((ext_vector_type(2))) float v2f;
typedef __attribute__((ext_vector_type(8))) float v8f;

#define WMMA_F32(a, b, c) \
  __builtin_amdgcn_wmma_f32_16x16x4_f32(false, (a), false, (b), (short)0, (c), false, false)

// ---------------------------------------------------------------------------
// Grid-wide generation barrier (all blocks resident: grid is only 64 WGs)
// ---------------------------------------------------------------------------
__device__ __forceinline__ void grid_barrier(int* cnt, int* gen, int nblocks) {
  __syncthreads();
  if (threadIdx.x == 0) {
    __threadfence();  // publish our stores (agent scope) before arriving
    int g = __hip_atomic_load(gen, __ATOMIC_RELAXED, __HIP_MEMORY_SCOPE_AGENT);
    int prev = __hip_atomic_fetch_add(cnt, 1, __ATOMIC_ACQ_REL, __HIP_MEMORY_SCOPE_AGENT);
    if (prev == nblocks - 1) {
      __hip_atomic_store(cnt, 0, __ATOMIC_RELAXED, __HIP_MEMORY_SCOPE_AGENT);
      __hip_atomic_fetch_add(gen, 1, __ATOMIC_ACQ_REL, __HIP_MEMORY_SCOPE_AGENT);
    } else {
      while (__hip_atomic_load(gen, __ATOMIC_ACQUIRE, __HIP_MEMORY_SCOPE_AGENT) == g) {
        __builtin_amdgcn_s_sleep(2);
      }
    }
  }
  __syncthreads();
}

// ---------------------------------------------------------------------------
// Zero the step accumulator + barrier control words
// ---------------------------------------------------------------------------
__global__ void __launch_bounds__(256) zero_ws(float* hwAcc, int* ctrl) {
  int t = blockIdx.x * 256 + threadIdx.x;
  if (t < B_ * HID_) hwAcc[t] = 0.0f;
  if (t < 8) ctrl[t] = 0;
}

// ---------------------------------------------------------------------------
// input_w[s,b,h] = sum_i u[b,s,i] * Wi[h,i] + bi[h]
// GEMM M=B*S(16384) N=256 K=64, one wave per 16x16 tile (f32 WMMA, K-step 4)
// ---------------------------------------------------------------------------
__global__ void __launch_bounds__(256) inproj_kernel(
    const float* __restrict__ u, const float* __restrict__ Wi,
    const float* __restrict__ bi, float* __restrict__ iw) {
  const int lane = threadIdx.x & 31;
  const int gw   = (blockIdx.x * blockDim.x + threadIdx.x) >> 5;  // 0..16383
  const int mt   = gw >> 4;            // 1024 M tiles
  const int nt   = gw & 15;            // 16 N tiles
  const int b    = mt >> 6;            // 64 M-tiles per batch (1024/16)
  const int s0   = (mt & 63) * 16;
  const int row  = lane & 15;
  const int koff = (lane >> 4) << 1;   // 0 or 2
  const float* arow = u  + ((size_t)b * S_ + s0 + row) * IN_ + koff;
  const float* bcol = Wi + (size_t)(nt * 16 + row) * IN_ + koff;  // row==n here
  v8f acc0 = {}, acc1 = {};
#pragma unroll
  for (int k = 0; k < IN_; k += 8) {
    v2f a0 = *(const v2f*)(arow + k);
    v2f b0 = *(const v2f*)(bcol + k);
    v2f a1 = *(const v2f*)(arow + k + 4);
    v2f b1 = *(const v2f*)(bcol + k + 4);
    acc0 = WMMA_F32(a0, b0, acc0);
    acc1 = WMMA_F32(a1, b1, acc1);
  }
  const int n  = lane & 15;
  const int m0 = (lane >> 4) * 8;
  const float bv = bi[nt * 16 + n];
#pragma unroll
  for (int r = 0; r < 8; ++r) {
    iw[(size_t)(s0 + m0 + r) * (B_ * HID_) + b * HID_ + nt * 16 + n] =
        acc0[r] + acc1[r] + bv;
  }
}

// ---------------------------------------------------------------------------
// Persistent scan kernel.  hw = G @ T3_flat^T with G[b,p] = h[b,i]*h[b,k],
// p = i*256+k.  64 blocks x 8 waves = 512 waves = 16 j-tiles x 32 k-chunks.
// ---------------------------------------------------------------------------
__global__ void __launch_bounds__(256, 1) rnn_scan_kernel(
    const float* __restrict__ hidden0, const float* __restrict__ T3,
    const float* __restrict__ bias,    const float* __restrict__ iw,
    float* __restrict__ hbuf, float* __restrict__ hwAcc, int* __restrict__ ctrl,
    float* __restrict__ r_out, float* __restrict__ hidden_final) {
  __shared__ float hs[B_ * HID_];  // 16 KB: current h staged per WGP

  const int lane = threadIdx.x & 31;
  const int gw   = blockIdx.x * 8 + (threadIdx.x >> 5);  // 0..511
  const int jt   = gw & 15;        // N tile (j)
  const int kc   = gw >> 4;        // 0..31 K chunk (2048 of 65536)
  const int row  = lane & 15;      // batch row for A / column j for B
  const int koff = (lane >> 4) << 1;
  const float* bcolBase = T3 + ((size_t)(jt * 16 + row) << 16) + koff;
  const int tid = blockIdx.x * 256 + threadIdx.x;
  const int n   = lane & 15;
  const int m0  = (lane >> 4) * 8;

  for (int s = 0; s < S_; ++s) {
    // ---- stage current h (16x256 f32) into LDS ----
    const float* hc = (s == 0) ? hidden0 : hbuf;
    for (int t = threadIdx.x; t < (B_ * HID_) / 4; t += 256)
      ((float4*)hs)[t] = ((const float4*)hc)[t];
    __syncthreads();

    // ---- partial GEMM: K range [kc*2048, kc*2048+2048) ----
    v8f acc0 = {}, acc1 = {};
    const float* hrow = &hs[row * HID_];
#pragma unroll 1
    for (int ii = 0; ii < 8; ++ii) {
      const int i = kc * 8 + ii;             // outer index of the h-outer-product
      const float hi = hrow[i];
      const float* bp = bcolBase + (size_t)i * HID_;
#pragma unroll 4
      for (int c = 0; c < HID_; c += 8) {
        v2f p0 = *(const v2f*)(hrow + c + koff);
        v2f p1 = *(const v2f*)(hrow + c + 4 + koff);
        v2f a0; a0.x = hi * p0.x; a0.y = hi * p0.y;
        v2f a1; a1.x = hi * p1.x; a1.y = hi * p1.y;
        v2f b0 = *(const v2f*)(bp + c);
        v2f b1 = *(const v2f*)(bp + c + 4);
        acc0 = WMMA_F32(a0, b0, acc0);
        acc1 = WMMA_F32(a1, b1, acc1);
      }
    }
    // ---- reduce partial 16x16 tile into L2-resident accumulator ----
#pragma unroll
    for (int r = 0; r < 8; ++r)
      atomicAdd(&hwAcc[(m0 + r) * HID_ + jt * 16 + n], acc0[r] + acc1[r]);

    grid_barrier(&ctrl[0], &ctrl[1], 64);

    // ---- state update: h_new = tanh(hw + bias + iw[s]) ----
    if (tid < B_ * HID_) {
      const int j = tid & (HID_ - 1);
      float v = hwAcc[tid];
      hwAcc[tid] = 0.0f;  // re-arm accumulator for next step
      v += bias[j] + iw[(size_t)s * (B_ * HID_) + tid];
      float hn = tanhf(v);
      hbuf[tid] = hn;
      const int b = tid >> 8;
      r_out[((size_t)b * S_ + s) * HID_ + j] = hn;
      if (s == S_ - 1) hidden_final[tid] = hn;
    }

    grid_barrier(&ctrl[0], &ctrl[1], 64);
  }
}

// ---------------------------------------------------------------------------
// output[b,s,o] = sum_h r_out[b,s,h] * Wo[o,h] + bo[o]
// GEMM M=16384 N=64 K=256, one wave per 16x16 tile
// ---------------------------------------------------------------------------
__global__ void __launch_bounds__(256) outproj_kernel(
    const float* __restrict__ r, const float* __restrict__ Wo,
    const float* __restrict__ bo, float* __restrict__ out) {
  const int lane = threadIdx.x & 31;
  const int gw   = (blockIdx.x * blockDim.x + threadIdx.x) >> 5;  // 0..4095
  const int mt   = gw >> 2;            // 1024 M tiles
  const int nt   = gw & 3;             // 4 N tiles
  const int row  = lane & 15;
  const int koff = (lane >> 4) << 1;
  const float* arow = r  + (size_t)(mt * 16 + row) * HID_ + koff;
  const float* bcol = Wo + (size_t)(nt * 16 + row) * HID_ + koff;
  v8f acc0 = {}, acc1 = {};
#pragma unroll 8
  for (int k = 0; k < HID_; k += 8) {
    v2f a0 = *(const v2f*)(arow + k);
    v2f b0 = *(const v2f*)(bcol + k);
    v2f a1 = *(const v2f*)(arow + k + 4);
    v2f b1 = *(const v2f*)(bcol + k + 4);
    acc0 = WMMA_F32(a0, b0, acc0);
    acc1 = WMMA_F32(a1, b1, acc1);
  }
  const int n  = lane & 15;
  const int m0 = (lane >> 4) * 8;
  const float bv = bo[nt * 16 + n];
#pragma unroll
  for (int r8 = 0; r8 < 8; ++r8)
    out[(size_t)(mt * 16 + m0 + r8) * OUT_ + nt * 16 + n] = acc0[r8] + acc1[r8] + bv;
}

// ---------------------------------------------------------------------------
extern "C" void kernel_launch(void* const* d_in, const int* in_sizes, int n_in,
                              void* d_out, int out_size, void* d_ws, size_t ws_size,
                              hipStream_t stream) {
  const float* u    = (const float*)d_in[0];  // [B,S,IN]
  const float* h0   = (const float*)d_in[1];  // [B,HID]
  const float* Wi   = (const float*)d_in[2];  // [HID,IN]
  const float* bi   = (const float*)d_in[3];  // [HID]
  const float* T3   = (const float*)d_in[4];  // [HID,HID,HID]
  const float* bias = (const float*)d_in[5];  // [HID]
  const float* Wo   = (const float*)d_in[6];  // [OUT,HID]
  const float* bo   = (const float*)d_in[7];  // [OUT]

  float* out_output = (float*)d_out;                       // [B,S,OUT]
  float* out_hidden = out_output + (size_t)B_ * S_ * OUT_; // [B,HID]
  float* out_rout   = out_hidden + (size_t)B_ * HID_;      // [B,S,HID]

  float* iw    = (float*)d_ws;                        // [S,B,HID] 16 MB
  float* hbuf  = iw    + (size_t)S_ * B_ * HID_;      // [B,HID]
  float* hwAcc = hbuf  + (size_t)B_ * HID_;           // [B,HID]
  int*   ctrl  = (int*)(hwAcc + (size_t)B_ * HID_);   // barrier cnt/gen

  // 1) arm the accumulator + barrier state
  zero_ws<<<17, 256, 0, stream>>>(hwAcc, ctrl);
  // 2) input projection: 16384 wave-tiles, 8 waves/block
  inproj_kernel<<<2048, 256, 0, stream>>>(u, Wi, bi, iw);
  // 3) persistent sequential scan (64 resident WGs, grid barriers per step)
  rnn_scan_kernel<<<64, 256, 0, stream>>>(h0, T3, bias, iw, hbuf, hwAcc, ctrl,
                                          out_rout, out_hidden);
  // 4) output projection: 4096 wave-tiles
  outproj_kernel<<<512, 256, 0, stream>>>(out_rout, Wo, bo, out_output);
}